// SoftCELoss_1219770712179
// MI455X (gfx1250) — compile-verified
//
#include <hip/hip_runtime.h>
#include <math.h>

// ---------------------------------------------------------------------------
// Soft partial-label cross-entropy, MI455X (gfx1250).
// Memory-bound: 256 MB streamed once -> ~11 us at 23.3 TB/s. Data path uses
// CDNA5 async global->LDS copies (GLOBAL_LOAD_ASYNC_TO_LDS_B128, ASYNCcnt),
// double-buffered one row (16 KB) per wave32. Stream is marked non-temporal
// (TH_LOAD_NT): 256 MB read-once > 192 MB L2, caching it is pure thrash.
// ---------------------------------------------------------------------------

#define C_DIM          4096          // classes (floats per row)
#define WAVES_PER_BLK  4
#define BLK_THREADS    (WAVES_PER_BLK * 32)
#define MAIN_GRID      1024
#define ALPHA_W        0.7f
// bytes per row = C_DIM*4 = 16384; each async b128 moves 32 lanes * 16B = 512B
#define ISSUES_PER_ROW (C_DIM * 4 / 512)   // 32

// --- CDNA5 async copy: one instruction copies 16B per lane, mem -> LDS.
// INST_OFFSET is added to BOTH the LDS address and the global address
// (CDNA5 ISA), so one immediate walks both sides.
template <int I>
struct AsyncIssue {
    __device__ static inline void run(unsigned lds_addr, unsigned long long gaddr) {
        AsyncIssue<I - 1>::run(lds_addr, gaddr);
        asm volatile("global_load_async_to_lds_b128 %0, %1, off offset:%2 th:TH_LOAD_NT"
                     :
                     : "v"(lds_addr), "v"(gaddr), "i"((I - 1) * 512)
                     : "memory");
    }
};
template <>
struct AsyncIssue<0> {
    __device__ static inline void run(unsigned, unsigned long long) {}
};

__device__ inline void wait_asynccnt_le(int keep) {
    if (keep == 0)
        asm volatile("s_wait_asynccnt 0" ::: "memory");
    else
        asm volatile("s_wait_asynccnt 32" ::: "memory");
}
__device__ inline void wait_dscnt0() {
    asm volatile("s_wait_dscnt 0" ::: "memory");
}

__global__ void __launch_bounds__(BLK_THREADS)
softce_main_kernel(const float* __restrict__ logits,
                   const int* __restrict__ plabels,
                   float* __restrict__ partials,   // [gridDim.x] when !atomic_mode
                   int B,
                   int atomic_mode,
                   float invB,
                   float* __restrict__ out_atomic) {
    extern __shared__ float sh[];                       // [WAVES][2][C_DIM] + wsum[WAVES]
    float* wsum = sh + WAVES_PER_BLK * 2 * C_DIM;

    const int lane = threadIdx.x & 31;
    const int w    = threadIdx.x >> 5;
    const int waveGid    = blockIdx.x * WAVES_PER_BLK + w;
    const int waveStride = gridDim.x * WAVES_PER_BLK;

    // LDS byte addresses of this wave's two row buffers (+ per-lane 16B slot).
    // Low 32 bits of a flat shared pointer are the LDS byte address.
    const unsigned lds0 = (unsigned)(size_t)(sh + (w * 2 + 0) * C_DIM) + (unsigned)lane * 16u;
    const unsigned lds1 = (unsigned)(size_t)(sh + (w * 2 + 1) * C_DIM) + (unsigned)lane * 16u;

    float acc = 0.0f;
    int   r   = waveGid;
    int   p   = 0;

    if (r < B) {
        unsigned long long g =
            (unsigned long long)(const void*)(logits + (size_t)r * C_DIM) + (unsigned)lane * 16u;
        AsyncIssue<ISSUES_PER_ROW>::run(lds0, g);
    }

    for (; r < B; r += waveStride) {
        const int  rn   = r + waveStride;
        const bool more = rn < B;

        // All prior LDS reads must retire before the TDM may overwrite a buffer.
        wait_dscnt0();
        if (more) {
            unsigned long long g =
                (unsigned long long)(const void*)(logits + (size_t)rn * C_DIM) + (unsigned)lane * 16u;
            AsyncIssue<ISSUES_PER_ROW>::run(p ? lds0 : lds1, g);
            wait_asynccnt_le(32);   // async loads complete in order: current buffer ready
        } else {
            wait_asynccnt_le(0);
        }

        const float4* bp  = (const float4*)(sh + (w * 2 + p) * C_DIM);
        const float*  row = (const float*)(sh + (w * 2 + p) * C_DIM);

        // Pass 1: row max.
        float m = -INFINITY;
#pragma unroll 8
        for (int k = 0; k < C_DIM / 128; ++k) {
            float4 v = bp[lane + k * 32];
            m = fmaxf(m, fmaxf(fmaxf(v.x, v.y), fmaxf(v.z, v.w)));
        }
        for (int off = 16; off > 0; off >>= 1)
            m = fmaxf(m, __shfl_xor(m, off, 32));

        // Pass 2: sum of exp(x - m).
        float s = 0.0f;
#pragma unroll 8
        for (int k = 0; k < C_DIM / 128; ++k) {
            float4 v = bp[lane + k * 32];
            s += __expf(v.x - m) + __expf(v.y - m) + __expf(v.z - m) + __expf(v.w - m);
        }
        for (int off = 16; off > 0; off >>= 1)
            s += __shfl_xor(s, off, 32);

        const float lse = m + __logf(s);

        if (lane == 0) {
            const int p0 = plabels[3 * r + 0];
            const int p1 = plabels[3 * r + 1];
            const int p2 = plabels[3 * r + 2];
            const float l0 = lse - row[p0];
            const float l1 = lse - row[p1];
            const float l2 = lse - row[p2];
            float per;
            if (p0 == p1)       per = ALPHA_W * (l0 + l1) * 0.5f + (1.0f - ALPHA_W) * l2;
            else if (p0 == p2)  per = ALPHA_W * (l0 + l2) * 0.5f + (1.0f - ALPHA_W) * l1;
            else if (p1 == p2)  per = ALPHA_W * (l1 + l2) * 0.5f + (1.0f - ALPHA_W) * l0;
            else                per = 0.0f;
            acc += per;
        }
        p ^= 1;
    }

    if (lane == 0) wsum[w] = acc;
    __syncthreads();
    if (threadIdx.x == 0) {
        float bs = 0.0f;
#pragma unroll
        for (int i = 0; i < WAVES_PER_BLK; ++i) bs += wsum[i];
        if (atomic_mode)
            atomicAdd(out_atomic, bs * invB);
        else
            partials[blockIdx.x] = bs;
    }
}

__global__ void softce_zero_kernel(float* out) {
    if (threadIdx.x == 0 && blockIdx.x == 0) *out = 0.0f;
}

__global__ void __launch_bounds__(256)
softce_reduce_kernel(const float* __restrict__ partials, int n,
                     float* __restrict__ out, float invB) {
    __shared__ float sm[256];
    float s = 0.0f;
    for (int i = threadIdx.x; i < n; i += 256) s += partials[i];
    sm[threadIdx.x] = s;
    __syncthreads();
    for (int off = 128; off > 0; off >>= 1) {
        if (threadIdx.x < off) sm[threadIdx.x] += sm[threadIdx.x + off];
        __syncthreads();
    }
    if (threadIdx.x == 0) *out = sm[0] * invB;
}

extern "C" void kernel_launch(void* const* d_in, const int* in_sizes, int n_in,
                              void* d_out, int out_size, void* d_ws, size_t ws_size,
                              hipStream_t stream) {
    const float* logits  = (const float*)d_in[0];
    // d_in[1] (targets) is unused by the reference math.
    const int*   plabels = (const int*)d_in[2];
    float*       out     = (float*)d_out;

    const int   B    = in_sizes[1];          // targets has B elements
    const float invB = 1.0f / (float)(B > 0 ? B : 1);

    int grid = (B + WAVES_PER_BLK - 1) / WAVES_PER_BLK;
    if (grid > MAIN_GRID) grid = MAIN_GRID;
    if (grid < 1) grid = 1;

    const size_t shmem = (size_t)(WAVES_PER_BLK * 2 * C_DIM + WAVES_PER_BLK) * sizeof(float);

    if (ws_size >= (size_t)grid * sizeof(float)) {
        // Deterministic two-stage reduction via workspace.
        softce_main_kernel<<<grid, BLK_THREADS, shmem, stream>>>(
            logits, plabels, (float*)d_ws, B, 0, invB, out);
        softce_reduce_kernel<<<1, 256, 0, stream>>>((const float*)d_ws, grid, out, invB);
    } else {
        // Fallback: zero the output then atomically accumulate scaled block sums.
        softce_zero_kernel<<<1, 32, 0, stream>>>(out);
        softce_main_kernel<<<grid, BLK_THREADS, shmem, stream>>>(
            logits, plabels, nullptr, B, 1, invB, out);
    }
}